// SimpleLSTM_16982300688545
// MI455X (gfx1250) — compile-verified
//
#include <hip/hip_runtime.h>
#include <hip/hip_bf16.h>
#include <math.h>

// ---------------------------------------------------------------------------
// SimpleLSTM for MI455X (gfx1250, wave32)
//   Phase 1: xz[t][g] = sum_d x[t][d]*W_ih[g][d] + b_ih[g] + b_hh[g]
//            fp32 WMMA GEMM (V_WMMA_F32_16X16X4_F32), M=T=65536, N=256, K=99.
//            x tile (128x99, 50.7KB) staged into LDS by the Tensor Data Mover
//            (tensor_load_to_lds + s_wait_tensorcnt), A fragments from LDS.
//   Phase 2: serial LSTM scan, single workgroup, W_hh rows in VGPRs,
//            h broadcast through LDS, 2 barriers/step. Head fused at end.
// ---------------------------------------------------------------------------

#define T_DIM   65536
#define IN_DIM  99
#define HID     64
#define GATES   256   // 4*HID
#define ROWS_PER_BLOCK 128

typedef __attribute__((ext_vector_type(2))) float        v2f;
typedef __attribute__((ext_vector_type(8))) float        v8f;
typedef __attribute__((ext_vector_type(4))) unsigned int v4u;
typedef __attribute__((ext_vector_type(4))) int          v4i;
typedef __attribute__((ext_vector_type(8))) int          v8i;

// ------------------------------ Phase 1 ------------------------------------
// grid: T/128 blocks of 256 threads (8 waves); each wave computes a 16x256
// tile of xz (16 f32 WMMA accumulators = 128 VGPRs).
__global__ __launch_bounds__(256) void lstm_xz_gemm(
    const float* __restrict__ x,      // [T][99]
    const float* __restrict__ W_ih,   // [256][99]
    const float* __restrict__ b_ih,   // [256]
    const float* __restrict__ b_hh,   // [256]
    float* __restrict__ xz)           // [T][256]
{
    __shared__ float xsh[ROWS_PER_BLOCK * IN_DIM];   // 50,688 B

    const int lane   = threadIdx.x & 31;
    const int wave   = threadIdx.x >> 5;
    const int rowblk = blockIdx.x * ROWS_PER_BLOCK;
    const int row0   = rowblk + wave * 16;           // first of this wave's 16 rows

    const int lr = lane & 15;   // row/col-in-tile selector
    const int hi = lane >> 4;   // high half of wave (K + 2)

    // --- TDM: DMA x[rowblk .. rowblk+127][0..98] into LDS (wave 0 only; the
    //     TDM ignores EXEC and TENSORcnt is per-wave, so one wave issues it).
    if (wave == 0) {
        const unsigned long long gaddr =
            (unsigned long long)(uintptr_t)x + (unsigned long long)rowblk * IN_DIM * 4ull;
        const unsigned int laddr = (unsigned int)(uintptr_t)&xsh[0]; // LDS byte offset

        v4u g0;
        g0[0] = 1u;                                   // count=1 (valid), no gather
        g0[1] = laddr;                                // lds_addr
        g0[2] = (unsigned int)gaddr;                  // global_addr[31:0]
        g0[3] = (unsigned int)((gaddr >> 32) & 0x1FFFFFFull)
                | 0x80000000u;                        // global_addr[56:32], type=2

        v8i g1;
        g1[0] = (int)(2u << 16);                      // data_size = 4B; mask=0
        g1[1] = (int)((unsigned)IN_DIM << 16);        // tensor_dim0 lo16 (=99)
        g1[2] = 0;                                    // dim0 hi16 | dim1 lo16 (65536->0)
        g1[3] = (int)(1u | ((unsigned)IN_DIM << 16)); // dim1 hi16 (=1) | tile_dim0=99
        g1[4] = ROWS_PER_BLOCK;                       // tile_dim1=128 | tile_dim2=0
        g1[5] = IN_DIM;                               // tensor_dim0_stride lo32 = 99
        g1[6] = 0;                                    // stride0 hi | stride1 lo
        g1[7] = 0;                                    // stride1 hi

        v4i g2 = (v4i){0, 0, 0, 0};
        v4i g3 = (v4i){0, 0, 0, 0};
#if __clang_major__ >= 23
        v8i g4 = (v8i){0, 0, 0, 0, 0, 0, 0, 0};
        __builtin_amdgcn_tensor_load_to_lds(g0, g1, g2, g3, g4, 0);
#else
        __builtin_amdgcn_tensor_load_to_lds(g0, g1, g2, g3, 0);
#endif
        __builtin_amdgcn_s_wait_tensorcnt(0);
    }
    __syncthreads();

    const float* xrow = &xsh[(wave * 16 + lr) * IN_DIM]; // this lane's A row (LDS)

    v8f acc[16];
#pragma unroll
    for (int nt = 0; nt < 16; ++nt)
        acc[nt] = v8f{0.f, 0.f, 0.f, 0.f, 0.f, 0.f, 0.f, 0.f};

    // ---- main K loop: kk = 0..92, every index < 99 -> unconditional loads
    for (int kk = 0; kk < 96; kk += 4) {
        const int k0 = kk + 2 * hi;
        v2f a;
        a.x = xrow[k0];
        a.y = xrow[k0 + 1];
#pragma unroll
        for (int nt = 0; nt < 16; ++nt) {
            const float* wrow = W_ih + (size_t)(nt * 16 + lr) * IN_DIM;
            v2f b;
            b.x = wrow[k0];
            b.y = wrow[k0 + 1];
            acc[nt] = __builtin_amdgcn_wmma_f32_16x16x4_f32(
                false, a, false, b, (short)0, acc[nt], false, false);
        }
    }

    // ---- peeled K tail (kk = 96): k = 96..98 valid, k = 99 zero-filled
    {
        const int k0 = 96 + 2 * hi;         // 96 (lo half) / 98 (hi half)
        v2f a;
        a.x = xrow[k0];
        a.y = hi ? 0.f : xrow[97];
#pragma unroll
        for (int nt = 0; nt < 16; ++nt) {
            const float* wrow = W_ih + (size_t)(nt * 16 + lr) * IN_DIM;
            v2f b;
            b.x = wrow[k0];
            b.y = hi ? 0.f : wrow[97];
            acc[nt] = __builtin_amdgcn_wmma_f32_16x16x4_f32(
                false, a, false, b, (short)0, acc[nt], false, false);
        }
    }

    // Store D + bias. Lane l, VGPR v holds D[m = v + 8*hi][n = lr].
#pragma unroll
    for (int nt = 0; nt < 16; ++nt) {
        const int col    = nt * 16 + lr;
        const float bias = b_ih[col] + b_hh[col];
#pragma unroll
        for (int v = 0; v < 8; ++v) {
            const int m = v + 8 * hi;
            xz[(size_t)(row0 + m) * GATES + col] = acc[nt][v] + bias;
        }
    }
}

// ------------------------------ Phase 2 ------------------------------------
// Single workgroup (256 threads = 8 wave32 on one WGP). Thread g owns gate
// row g: its 64-entry W_hh row lives entirely in VGPRs. h (64 floats) is
// broadcast via LDS. c[j] lives in the registers of thread j (j < 64).
__global__ __launch_bounds__(256) void lstm_scan(
    const float* __restrict__ xz,     // [T][256]
    const float* __restrict__ W_hh,   // [256][64]
    const float* __restrict__ W1,     // [32][64]
    const float* __restrict__ W2,     // [3][32]
    const float* __restrict__ b2,     // [3]
    float* __restrict__ out)          // [3]
{
    __shared__ float h_sh[HID];
    __shared__ float gates_sh[GATES];
    __shared__ float y1_sh[32];

    const int g = threadIdx.x;
    const int gtype = g >> 6;         // 0:i 1:f 2:g 3:o

    // Preload W_hh row g into registers (64 VGPRs / thread).
    float w[HID];
#pragma unroll
    for (int j = 0; j < HID; j += 4) {
        const float4 v4 = *reinterpret_cast<const float4*>(W_hh + (size_t)g * HID + j);
        w[j] = v4.x; w[j + 1] = v4.y; w[j + 2] = v4.z; w[j + 3] = v4.w;
    }

    if (g < HID) h_sh[g] = 0.f;
    float c = 0.f;
    __syncthreads();

    for (int t = 0; t < T_DIM; ++t) {
        // Prefetch the xz row a few steps ahead (global_prefetch_b8).
        if (t + 8 < T_DIM)
            __builtin_prefetch(&xz[(size_t)(t + 8) * GATES + g], 0, 0);

        // z[g] = xz[t][g] + dot(W_hh[g,:], h)
        float acc = xz[(size_t)t * GATES + g];
#pragma unroll
        for (int j = 0; j < HID; ++j) acc += w[j] * h_sh[j];

        // gate nonlinearity (TRANS pipe: v_exp / tanh)
        const float act = (gtype == 2) ? tanhf(acc)
                                       : 1.f / (1.f + __expf(-acc));
        gates_sh[g] = act;
        __syncthreads();

        if (g < HID) {
            const float iv = gates_sh[g];
            const float fv = gates_sh[HID + g];
            const float gv = gates_sh[2 * HID + g];
            const float ov = gates_sh[3 * HID + g];
            c = fv * c + iv * gv;
            h_sh[g] = ov * tanhf(c);
        }
        __syncthreads();
    }

    // --------- MLP head: relu -> W1(32x64) -> relu -> W2(3x32)+b2 ----------
    if (g < 32) {
        float a = 0.f;
#pragma unroll
        for (int j = 0; j < HID; ++j)
            a += W1[(size_t)g * HID + j] * fmaxf(h_sh[j], 0.f);
        y1_sh[g] = fmaxf(a, 0.f);
    }
    __syncthreads();
    if (g < 3) {
        float a = b2[g];
#pragma unroll
        for (int i = 0; i < 32; ++i) a += W2[(size_t)g * 32 + i] * y1_sh[i];
        out[g] = a;
    }
}

// ------------------------------ Launch -------------------------------------
extern "C" void kernel_launch(void* const* d_in, const int* in_sizes, int n_in,
                              void* d_out, int out_size, void* d_ws, size_t ws_size,
                              hipStream_t stream) {
    const float* x    = (const float*)d_in[0];   // [1,T,99]
    const float* W_ih = (const float*)d_in[1];   // [256,99]
    const float* W_hh = (const float*)d_in[2];   // [256,64]
    const float* b_ih = (const float*)d_in[3];   // [256]
    const float* b_hh = (const float*)d_in[4];   // [256]
    const float* W1   = (const float*)d_in[5];   // [32,64]
    const float* W2   = (const float*)d_in[6];   // [3,32]
    const float* b2   = (const float*)d_in[7];   // [3]
    float* out = (float*)d_out;                  // [3]
    float* xz  = (float*)d_ws;                   // [T][256] scratch (64 MB)

    lstm_xz_gemm<<<T_DIM / ROWS_PER_BLOCK, 256, 0, stream>>>(x, W_ih, b_ih, b_hh, xz);
    lstm_scan<<<1, 256, 0, stream>>>(xz, W_hh, W1, W2, b2, out);
}